// EMDLoss_42348377538968
// MI455X (gfx1250) — compile-verified
//
#include <hip/hip_runtime.h>
#include <cmath>

// ---------------------------------------------------------------------------
// Entropic EMD (Sinkhorn) for B=8, N=M=2048, 3-D points, eps=0.005, 50 iters.
//
// f_i = eps*(log_mu - LSE_j((g_j - C_ij)/eps)),  g symmetric; C never stored.
// C_ij = r_i + s_j - 2 p_i.q_j ; the -2 p.q term is computed 16x16 tiles at a
// time with V_WMMA_F32_16X16X4_F32 (K=3 padded to 4). All LSE math is done in
// the log2 domain so the transcendental path is bare v_exp_f32 / v_log_f32.
// A-operand panels are pre-swizzled into LDS in the exact WMMA lane layout so
// the inner loop is one ds_load_b64 + one wmma + exp2 chain, branch-free.
// ---------------------------------------------------------------------------

typedef __attribute__((ext_vector_type(2))) float v2f;
typedef __attribute__((ext_vector_type(8))) float v8f;

static __device__ __forceinline__ float fast_exp2(float x) {
  return __builtin_amdgcn_exp2f(x);   // v_exp_f32 (base-2)
}
static __device__ __forceinline__ float fast_log2(float x) {
  return __builtin_amdgcn_logf(x);    // v_log_f32 (base-2)
}

namespace {
constexpr int   kB       = 8;
constexpr int   kN       = 2048;
constexpr float kEps     = 0.005f;
constexpr float kLn2     = 0.6931471805599453f;
constexpr float kInvEps2 = 1.0f / (kEps * kLn2);   // (x)/eps in log2 domain
constexpr float kAScale  = 2.0f * kInvEps2;        // folds "-2 p.q" scaling into WMMA A
constexpr float kEpsLn2  = kEps * kLn2;            // eps*ln2 (log2 -> cost units)
constexpr float kLogMu   = -7.624618986159398f;    // -ln(2048) = log_mu = log_nu
constexpr int   kITiles  = kN / 16;                // 128 row tiles per batch
constexpr int   kWaves   = 8;                      // 256-thread block = 8 wave32
constexpr int   kTilesPW = kITiles / kWaves;       // 16 j-tiles per wave
constexpr int   kAPairs  = kITiles * 32;           // 4096 (tile, lane) A-panel pairs
}

// Stage the WMMA A-operand panel for all 128 column tiles into LDS, already in
// the 16x4 f32 A layout: lane l<16 -> (q.x, q.y)*scale for point jbase+l,
// lane l>=16 -> (q.z*scale, 0) for point jbase+(l-16).
static __device__ __forceinline__ void stage_a_panel(
    float2* lds_a, const float* __restrict__ colPts, int b) {
  for (int idx = threadIdx.x; idx < kAPairs; idx += 256) {
    const int tile = idx >> 5;
    const int l    = idx & 31;
    const int j    = tile * 16 + (idx & 15);
    const float* qp = &colPts[(b * kN + j) * 3];
    float2 v;
    if (l < 16) { v.x = qp[0] * kAScale; v.y = qp[1] * kAScale; }
    else        { v.x = qp[2] * kAScale; v.y = 0.0f; }
    lds_a[idx] = v;
  }
}

// r[b,i] = |p_i|^2, s[b,j] = |q_j|^2, g = 0
__global__ __launch_bounds__(256) void emd_init(
    const float* __restrict__ preds, const float* __restrict__ gts,
    float* __restrict__ r, float* __restrict__ s, float* __restrict__ g) {
  const int idx = blockIdx.x * 256 + threadIdx.x;
  if (idx < kB * kN) {
    const float px = preds[idx * 3 + 0], py = preds[idx * 3 + 1], pz = preds[idx * 3 + 2];
    r[idx] = px * px + py * py + pz * pz;
    const float qx = gts[idx * 3 + 0], qy = gts[idx * 3 + 1], qz = gts[idx * 3 + 2];
    s[idx] = qx * qx + qy * qy + qz * qz;
    g[idx] = 0.0f;
  }
}

// One half-iteration: dualOut_i = eps*(log_mu - LSE_j((dualIn_j - C_ij)/eps)).
// Block: one 16-row tile of one batch. Waves split the 2048-column range.
__global__ __launch_bounds__(256) void emd_dual_update(
    const float* __restrict__ rowPts, const float* __restrict__ colPts,
    const float* __restrict__ rowNorm, const float* __restrict__ colNorm,
    const float* __restrict__ dualIn, float* __restrict__ dualOut) {
  __shared__ float2 lds_a[kAPairs];        // 32 KB pre-swizzled A panel
  __shared__ float  lds_gs[kN];            // (dualIn_j - s_j)/(eps*ln2)
  __shared__ float  red_m[kWaves][16];
  __shared__ float  red_s[kWaves][16];

  const int b = blockIdx.x >> 7;           // 128 row tiles per batch
  const int itile = blockIdx.x & 127;

  stage_a_panel(lds_a, colPts, b);
  for (int k = threadIdx.x; k < kN; k += 256)
    lds_gs[k] = (dualIn[b * kN + k] - colNorm[b * kN + k]) * kInvEps2;
  __syncthreads();

  const int lane = threadIdx.x & 31;
  const int wave = threadIdx.x >> 5;
  const int half = lane >> 4;
  const int m16  = lane & 15;
  const int i    = itile * 16 + m16;

  const float ri = rowNorm[b * kN + i] * kInvEps2;
  const float px = rowPts[(b * kN + i) * 3 + 0];
  const float py = rowPts[(b * kN + i) * 3 + 1];
  const float pz = rowPts[(b * kN + i) * 3 + 2];
  // B operand (4x16, rows striped across lanes): K0/K1 rows in lanes 0-15,
  // K2/K3(pad) rows in lanes 16-31.
  v2f bmat;
  bmat[0] = half ? pz : px;
  bmat[1] = half ? 0.0f : py;

  float m = -1e30f, ssum = 0.0f;           // online LSE state (log2 domain)

#pragma unroll 4
  for (int t = 0; t < kTilesPW; ++t) {
    const int tile  = wave * kTilesPW + t;
    const int jbase = tile * 16;
    const float2 a2 = lds_a[tile * 32 + lane];   // one ds_load_b64, branch-free
    v2f amat; amat[0] = a2.x; amat[1] = a2.y;
    v8f c = {};
    // D[M=j][N=i] tile of (2/(eps*ln2)) * q_j . p_i
    v8f d = __builtin_amdgcn_wmma_f32_16x16x4_f32(false, amat, false, bmat,
                                                  (short)0, c, false, false);
    const float4 g0 = *(const float4*)&lds_gs[jbase + half * 8];
    const float4 g1 = *(const float4*)&lds_gs[jbase + half * 8 + 4];
    const float a0 = d[0] + g0.x - ri;
    const float a1 = d[1] + g0.y - ri;
    const float a2s = d[2] + g0.z - ri;
    const float a3 = d[3] + g0.w - ri;
    const float a4 = d[4] + g1.x - ri;
    const float a5 = d[5] + g1.y - ri;
    const float a6 = d[6] + g1.z - ri;
    const float a7 = d[7] + g1.w - ri;
    const float tm = fmaxf(fmaxf(fmaxf(a0, a1), fmaxf(a2s, a3)),
                           fmaxf(fmaxf(a4, a5), fmaxf(a6, a7)));
    const float nm = fmaxf(m, tm);
    const float e = (fast_exp2(a0 - nm) + fast_exp2(a1 - nm)) +
                    (fast_exp2(a2s - nm) + fast_exp2(a3 - nm)) +
                    (fast_exp2(a4 - nm) + fast_exp2(a5 - nm)) +
                    (fast_exp2(a6 - nm) + fast_exp2(a7 - nm));
    ssum = ssum * fast_exp2(m - nm) + e;
    m = nm;
  }

  // merge the two half-wave partials (same row i, disjoint j sets)
  {
    const float om = __shfl_xor(m, 16, 32);
    const float os = __shfl_xor(ssum, 16, 32);
    const float nm = fmaxf(m, om);
    ssum = ssum * fast_exp2(m - nm) + os * fast_exp2(om - nm);
    m = nm;
  }
  if (half == 0) { red_m[wave][m16] = m; red_s[wave][m16] = ssum; }
  __syncthreads();

  if (threadIdx.x < 16) {
    float M = -1e30f, S = 0.0f;
    for (int w = 0; w < kWaves; ++w) {
      const float mw = red_m[w][threadIdx.x];
      const float sw = red_s[w][threadIdx.x];
      const float nm = fmaxf(M, mw);
      S = S * fast_exp2(M - nm) + sw * fast_exp2(mw - nm);
      M = nm;
    }
    const float lse = kLn2 * (M + fast_log2(S));   // back to natural-log LSE
    dualOut[b * kN + itile * 16 + threadIdx.x] = kEps * (kLogMu - lse);
  }
}

// loss partials: per block, sum_{i,j in tile} exp((f+g-C)/eps) * C
__global__ __launch_bounds__(256) void emd_loss(
    const float* __restrict__ rowPts, const float* __restrict__ colPts,
    const float* __restrict__ rowNorm, const float* __restrict__ colNorm,
    const float* __restrict__ f, const float* __restrict__ g,
    float* __restrict__ partial) {
  __shared__ float2 lds_a[kAPairs];        // 32 KB pre-swizzled A panel
  __shared__ float  lds_gs[kN];            // (g_j - s_j)/(eps*ln2)
  __shared__ float  lds_g[kN];             // raw g_j (to reconstruct C)
  __shared__ float  redw[kWaves];

  const int b = blockIdx.x >> 7;
  const int itile = blockIdx.x & 127;

  stage_a_panel(lds_a, colPts, b);
  for (int k = threadIdx.x; k < kN; k += 256) {
    const float gk = g[b * kN + k];
    lds_g[k] = gk;
    lds_gs[k] = (gk - colNorm[b * kN + k]) * kInvEps2;
  }
  __syncthreads();

  const int lane = threadIdx.x & 31;
  const int wave = threadIdx.x >> 5;
  const int half = lane >> 4;
  const int m16  = lane & 15;
  const int i    = itile * 16 + m16;

  const float fi  = f[b * kN + i];
  const float fri = (fi - rowNorm[b * kN + i]) * kInvEps2;  // (f_i - r_i)/(eps*ln2)
  const float px = rowPts[(b * kN + i) * 3 + 0];
  const float py = rowPts[(b * kN + i) * 3 + 1];
  const float pz = rowPts[(b * kN + i) * 3 + 2];
  v2f bmat;
  bmat[0] = half ? pz : px;
  bmat[1] = half ? 0.0f : py;

  float acc = 0.0f;
#pragma unroll 4
  for (int t = 0; t < kTilesPW; ++t) {
    const int tile  = wave * kTilesPW + t;
    const int jbase = tile * 16;
    const float2 a2 = lds_a[tile * 32 + lane];
    v2f amat; amat[0] = a2.x; amat[1] = a2.y;
    v8f c = {};
    v8f d = __builtin_amdgcn_wmma_f32_16x16x4_f32(false, amat, false, bmat,
                                                  (short)0, c, false, false);
    const float4 gs0 = *(const float4*)&lds_gs[jbase + half * 8];
    const float4 gs1 = *(const float4*)&lds_gs[jbase + half * 8 + 4];
    const float4 gg0 = *(const float4*)&lds_g[jbase + half * 8];
    const float4 gg1 = *(const float4*)&lds_g[jbase + half * 8 + 4];
    const float gsv[8] = {gs0.x, gs0.y, gs0.z, gs0.w, gs1.x, gs1.y, gs1.z, gs1.w};
    const float ggv[8] = {gg0.x, gg0.y, gg0.z, gg0.w, gg1.x, gg1.y, gg1.z, gg1.w};
#pragma unroll
    for (int k = 0; k < 8; ++k) {
      const float arg2 = d[k] + gsv[k] + fri;        // log2 of P_ij
      const float P = fast_exp2(arg2);
      const float C = fi + ggv[k] - kEpsLn2 * arg2;  // C = f+g - eps*ln(P)
      acc += P * C;
    }
  }

  // deterministic block reduction: wave shuffle tree, then LDS across waves
  for (int off = 16; off > 0; off >>= 1) acc += __shfl_xor(acc, off, 32);
  if (lane == 0) redw[wave] = acc;
  __syncthreads();
  if (threadIdx.x == 0) {
    float t = 0.0f;
    for (int w = 0; w < kWaves; ++w) t += redw[w];
    partial[blockIdx.x] = t;
  }
}

__global__ __launch_bounds__(256) void emd_reduce(
    const float* __restrict__ partial, float* __restrict__ out) {
  __shared__ float buf[256];
  float a = 0.0f;
  for (int k = threadIdx.x; k < kB * kITiles; k += 256) a += partial[k];
  buf[threadIdx.x] = a;
  __syncthreads();
  for (int s = 128; s > 0; s >>= 1) {
    if (threadIdx.x < s) buf[threadIdx.x] += buf[threadIdx.x + s];
    __syncthreads();
  }
  if (threadIdx.x == 0) out[0] = buf[0];
}

extern "C" void kernel_launch(void* const* d_in, const int* in_sizes, int n_in,
                              void* d_out, int out_size, void* d_ws, size_t ws_size,
                              hipStream_t stream) {
  (void)in_sizes; (void)n_in; (void)out_size; (void)ws_size;
  const float* preds = (const float*)d_in[0];
  const float* gts   = (const float*)d_in[1];
  float* out = (float*)d_out;

  float* ws = (float*)d_ws;
  float* r = ws;                 // [B*N] |p|^2
  float* s = r + kB * kN;        // [B*N] |q|^2
  float* f = s + kB * kN;        // [B*N] source dual
  float* g = f + kB * kN;        // [B*N] target dual
  float* partial = g + kB * kN;  // [1024] block partial losses

  emd_init<<<(kB * kN) / 256, 256, 0, stream>>>(preds, gts, r, s, g);

  const int nblk = kB * kITiles;  // 1024
  for (int it = 0; it < 50; ++it) {
    // f-update: rows = preds(i), cols = gts(j), uses g
    emd_dual_update<<<nblk, 256, 0, stream>>>(preds, gts, r, s, g, f);
    // g-update: rows = gts(j), cols = preds(i), uses new f
    emd_dual_update<<<nblk, 256, 0, stream>>>(gts, preds, s, r, f, g);
  }
  emd_loss<<<nblk, 256, 0, stream>>>(preds, gts, r, s, f, g, partial);
  emd_reduce<<<1, 256, 0, stream>>>(partial, out);
}